// DyFraudNet_53266184405473
// MI455X (gfx1250) — compile-verified
//
#include <hip/hip_runtime.h>
#include <math.h>

#define SLOPE 0.01f

typedef float v2f __attribute__((ext_vector_type(2)));
typedef float v8f __attribute__((ext_vector_type(8)));

__device__ __forceinline__ float leaky(float x) { return x >= 0.0f ? x : SLOPE * x; }

// ---------------------------------------------------------------------------
// Fused preprocess: h = leaky(leaky(x @ W1^T + b1) @ W2^T + b2)
// One block = 16 node rows. The 16x128 x-tile (contiguous 8 KB) is staged into
// LDS with GLOBAL_LOAD_ASYNC_TO_LDS_B128 (ASYNCcnt path). 8 waves each compute
// two 16x16 col-tiles of the 256-wide hidden layer with
// V_WMMA_F32_16X16X4_F32 (fp32-exact), stash the leaky()'d 16x256 tile in LDS,
// then wave 0 contracts K=256 down to the 16 outputs.
// ---------------------------------------------------------------------------
__global__ __launch_bounds__(256) void k_pre(const float* __restrict__ x,
                                             const float* __restrict__ W1,
                                             const float* __restrict__ b1,
                                             const float* __restrict__ W2,
                                             const float* __restrict__ b2,
                                             float* __restrict__ hA, int N) {
  __shared__ float Xs[16][128];   // 8 KB, one contiguous x tile
  __shared__ float Hs[16][256];   // 16 KB
  const int tid  = threadIdx.x;
  const int row0 = blockIdx.x * 16;

  if (row0 + 16 <= N) {
    // async Global -> LDS: 512 x B128 chunks, 2 per thread, no VGPR staging
    const unsigned xs_base = (unsigned)(uintptr_t)(&Xs[0][0]);
    const uint64_t ga_base = (uint64_t)(uintptr_t)(x + (size_t)row0 * 128);
    for (int c = tid; c < 512; c += 256) {
      unsigned lds = xs_base + (unsigned)c * 16u;
      uint64_t ga  = ga_base + (uint64_t)c * 16u;
      asm volatile("global_load_async_to_lds_b128 %0, %1, off"
                   :: "v"(lds), "v"(ga) : "memory");
    }
    asm volatile("s_wait_asynccnt 0x0" ::: "memory");
  } else {
    // partial last tile: scalar staging with zero padding
    for (int i = tid; i < 16 * 128; i += 256) {
      int r = i >> 7, k = i & 127;
      Xs[r][k] = (row0 + r < N) ? x[(size_t)(row0 + r) * 128 + k] : 0.0f;
    }
  }
  __syncthreads();

  const int wave = tid >> 5, lane = tid & 31;
  const int m = lane & 15, half = lane >> 4;

  // ---- stage 1: x(16x128) @ W1^T -> 16x256 tile ----
  for (int ct = 0; ct < 2; ++ct) {
    const int col = (wave * 2 + ct) * 16;
    v8f c;
    const float bias = b1[col + m];
#pragma unroll
    for (int g = 0; g < 8; ++g) c[g] = bias;
    for (int k0 = 0; k0 < 128; k0 += 4) {
      v2f a, b;
      a.x = Xs[m][k0 + 2 * half];
      a.y = Xs[m][k0 + 2 * half + 1];
      const float* wp = &W1[(size_t)(col + m) * 128 + k0 + 2 * half];  // B[k][n]=W1[n][k]
      b.x = wp[0];
      b.y = wp[1];
      c = __builtin_amdgcn_wmma_f32_16x16x4_f32(false, a, false, b, (short)0, c, false, false);
    }
#pragma unroll
    for (int g = 0; g < 8; ++g) Hs[g + 8 * half][col + m] = leaky(c[g]);
  }
  __syncthreads();

  // ---- stage 2: tile(16x256) @ W2^T -> 16x16, wave 0 only ----
  if (wave == 0) {
    v8f c;
    const float bias = b2[m];
#pragma unroll
    for (int g = 0; g < 8; ++g) c[g] = bias;
    for (int k0 = 0; k0 < 256; k0 += 4) {
      v2f a, b;
      a.x = Hs[m][k0 + 2 * half];
      a.y = Hs[m][k0 + 2 * half + 1];
      const float* wp = &W2[(size_t)m * 256 + k0 + 2 * half];  // B[k][n]=W2[n][k]
      b.x = wp[0];
      b.y = wp[1];
      c = __builtin_amdgcn_wmma_f32_16x16x4_f32(false, a, false, b, (short)0, c, false, false);
    }
#pragma unroll
    for (int g = 0; g < 8; ++g) {
      int r = row0 + g + 8 * half;
      if (r < N) hA[(size_t)r * 16 + m] = leaky(c[g]);
    }
  }
}

// ---------------------------------------------------------------------------
// xl = h(N x 16) @ Wnew^T (16x16). 8 waves/block, 16 rows per wave (WMMA K=16).
// ---------------------------------------------------------------------------
__global__ __launch_bounds__(256) void k_xl(const float* __restrict__ h,
                                            const float* __restrict__ Wnew,
                                            float* __restrict__ xl, int N) {
  const int tid = threadIdx.x;
  const int wave = tid >> 5, lane = tid & 31;
  const int m = lane & 15, half = lane >> 4;
  const int base = blockIdx.x * 128 + wave * 16;
  if (base >= N) return;  // wave-uniform

  v8f c = {};
#pragma unroll
  for (int k0 = 0; k0 < 16; k0 += 4) {
    v2f a, b;
    const int r = base + m;
    a.x = (r < N) ? h[(size_t)r * 16 + k0 + 2 * half] : 0.0f;
    a.y = (r < N) ? h[(size_t)r * 16 + k0 + 2 * half + 1] : 0.0f;
    b.x = Wnew[m * 16 + k0 + 2 * half];      // B[k][n]=Wnew[n][k]
    b.y = Wnew[m * 16 + k0 + 2 * half + 1];
    c = __builtin_amdgcn_wmma_f32_16x16x4_f32(false, a, false, b, (short)0, c, false, false);
  }
#pragma unroll
  for (int g = 0; g < 8; ++g) {
    int r = base + g + 8 * half;
    if (r < N) xl[(size_t)r * 16 + m] = c[g];
  }
}

// ---------------------------------------------------------------------------
// Tiny GRU + weight-transform for both layers, plus fused output head params.
// h0 == 0 in the reference, so gh = bhh and z*h == 0.
// params layout: [0,256) Wnew1, [256,512) Wnew2, [512,528) wsum, [528] bsum
// ---------------------------------------------------------------------------
__global__ __launch_bounds__(256) void k_params(
    const float* mem1, const float* Wih1, const float* bih1, const float* bhh1,
    const float* wtW1, const float* wtb1,
    const float* mem2, const float* Wih2, const float* bih2, const float* bhh2,
    const float* wtW2, const float* wtb2,
    const float* Wout, const float* bout, float* params) {
  __shared__ float s_gi[48];
  __shared__ float s_mem[16];
  const int t = threadIdx.x;
  for (int l = 0; l < 2; ++l) {
    const float* mem = l ? mem2 : mem1;
    const float* Wih = l ? Wih2 : Wih1;
    const float* bih = l ? bih2 : bih1;
    const float* bhh = l ? bhh2 : bhh1;
    const float* wtW = l ? wtW2 : wtW1;
    const float* wtb = l ? wtb2 : wtb1;
    if (t < 48) {
      float g = bih[t];
      for (int k = 0; k < 16; ++k) g += Wih[t * 16 + k] * mem[k];
      s_gi[t] = g;
    }
    __syncthreads();
    if (t < 16) {
      float r  = 1.0f / (1.0f + expf(-(s_gi[t] + bhh[t])));
      float z  = 1.0f / (1.0f + expf(-(s_gi[16 + t] + bhh[16 + t])));
      float nn = tanhf(s_gi[32 + t] + r * bhh[32 + t]);
      s_mem[t] = (1.0f - z) * nn;  // + z*h0, h0 == 0
    }
    __syncthreads();
    {
      float w = wtb[t];
      for (int k = 0; k < 16; ++k) w += wtW[t * 16 + k] * s_mem[k];
      params[l * 256 + t] = w;
    }
    __syncthreads();
  }
  if (t < 16) params[512 + t] = Wout[t] + Wout[16 + t];
  if (t == 0) params[528] = bout[0] + bout[1];
}

// ---------------------------------------------------------------------------
__global__ void k_zero(float* __restrict__ p, size_t n) {
  size_t i = (size_t)blockIdx.x * blockDim.x + threadIdx.x;
  if (i < n) p[i] = 0.0f;
}

__global__ void k_deg(const int* __restrict__ col, float* __restrict__ deg, int E, int N) {
  int i = blockIdx.x * 256 + threadIdx.x;
  if (i >= E + N) return;
  int c = (i < E) ? col[i] : (i - E);  // self-loops appended
  atomicAdd(&deg[c], 1.0f);
}

__global__ void k_dis(float* __restrict__ deg, int N) {
  int i = blockIdx.x * 256 + threadIdx.x;
  if (i >= N) return;
  float d = deg[i];
  deg[i] = d > 0.0f ? rsqrtf(d) : 0.0f;  // in-place: deg becomes dis
}

// one lane per (edge, feature): coalesced 16-float groups, f32 global atomics
__global__ void k_scatter(const int* __restrict__ row, const int* __restrict__ col,
                          const float* __restrict__ dis, const float* __restrict__ xl,
                          float* __restrict__ agg, int E, int N) {
  long long idx = (long long)blockIdx.x * 256 + threadIdx.x;
  long long total = (long long)(E + N) * 16;
  if (idx >= total) return;
  int e = (int)(idx >> 4), j = (int)(idx & 15);
  int r, c;
  if (e < E) { r = row[e]; c = col[e]; } else { r = c = e - E; }
  float w = dis[r] * dis[c];
  atomicAdd(&agg[(size_t)c * 16 + j], xl[(size_t)r * 16 + j] * w);
}

__global__ void k_finish(const float* __restrict__ agg, const float* __restrict__ bias,
                         float* __restrict__ hout, size_t n16) {
  size_t i = (size_t)blockIdx.x * 256 + threadIdx.x;
  if (i >= n16) return;
  hout[i] = leaky(agg[i] + bias[i & 15]);
}

// head: [0,16) wsum = Wout[0]+Wout[1], [16] bsum = bout[0]+bout[1]
__global__ void k_final(const float* __restrict__ agg, const float* __restrict__ bias,
                        const float* __restrict__ head, float* __restrict__ out, int N) {
  int n = blockIdx.x * 256 + threadIdx.x;
  if (n >= N) return;
  float acc = head[16];
#pragma unroll
  for (int j = 0; j < 16; ++j)
    acc += leaky(agg[(size_t)n * 16 + j] + bias[j]) * head[j];
  out[n] = acc;
}

// ---------------------------------------------------------------------------
extern "C" void kernel_launch(void* const* d_in, const int* in_sizes, int n_in,
                              void* d_out, int out_size, void* d_ws, size_t ws_size,
                              hipStream_t stream) {
  const float* x      = (const float*)d_in[0];
  const int*   ei     = (const int*)d_in[1];
  const float* W1     = (const float*)d_in[2];
  const float* b1     = (const float*)d_in[3];
  const float* W2     = (const float*)d_in[4];
  const float* b2     = (const float*)d_in[5];
  const float* mem1   = (const float*)d_in[6];
  const float* g1_Wih = (const float*)d_in[7];
  const float* g1_bih = (const float*)d_in[9];
  const float* g1_bhh = (const float*)d_in[10];
  const float* wt1_W  = (const float*)d_in[11];
  const float* wt1_b  = (const float*)d_in[12];
  const float* gcn1_b = (const float*)d_in[13];
  const float* mem2   = (const float*)d_in[14];
  const float* g2_Wih = (const float*)d_in[15];
  const float* g2_bih = (const float*)d_in[17];
  const float* g2_bhh = (const float*)d_in[18];
  const float* wt2_W  = (const float*)d_in[19];
  const float* wt2_b  = (const float*)d_in[20];
  const float* gcn2_b = (const float*)d_in[21];
  const float* Wout   = (const float*)d_in[22];
  const float* bout   = (const float*)d_in[23];

  const int N = in_sizes[0] / 128;
  const int E = in_sizes[1] / 2;
  const int* row = ei;
  const int* col = ei + E;

  float* ws     = (float*)d_ws;
  float* params = ws;                        // 1024 floats reserved
  float* hA     = ws + 1024;                 // N*16
  float* hB     = hA + (size_t)N * 16;       // N*16
  float* xl     = hB + (size_t)N * 16;       // N*16
  float* agg    = xl + (size_t)N * 16;       // N*16
  float* deg    = agg + (size_t)N * 16;      // N   (becomes dis in-place)

  const size_t n16 = (size_t)N * 16;
  const unsigned gN    = (N + 255) / 256;
  const unsigned gEN   = (E + N + 255) / 256;
  const unsigned gN16  = (unsigned)((n16 + 255) / 256);
  const unsigned gScat = (unsigned)(((long long)(E + N) * 16 + 255) / 256);
  const unsigned gPre  = (N + 15) / 16;
  const unsigned gXl   = (N + 127) / 128;

  // tiny GRU + regenerated GCN weights + fused output head
  k_params<<<1, 256, 0, stream>>>(mem1, g1_Wih, g1_bih, g1_bhh, wt1_W, wt1_b,
                                  mem2, g2_Wih, g2_bih, g2_bhh, wt2_W, wt2_b,
                                  Wout, bout, params);
  // degree -> dis (shared by both GCN layers)
  k_zero<<<gN, 256, 0, stream>>>(deg, (size_t)N);
  k_deg<<<gEN, 256, 0, stream>>>(col, deg, E, N);
  k_dis<<<gN, 256, 0, stream>>>(deg, N);
  // fused 128->256->16 MLP (WMMA + async Global->LDS staging)
  k_pre<<<gPre, 256, 0, stream>>>(x, W1, b1, W2, b2, hA, N);

  // ---- EvolveGCN layer 1 ----
  k_zero<<<gN16, 256, 0, stream>>>(agg, n16);
  k_xl<<<gXl, 256, 0, stream>>>(hA, params + 0, xl, N);
  k_scatter<<<gScat, 256, 0, stream>>>(row, col, deg, xl, agg, E, N);
  k_finish<<<gN16, 256, 0, stream>>>(agg, gcn1_b, hB, n16);

  // ---- EvolveGCN layer 2 + fused output head ----
  k_zero<<<gN16, 256, 0, stream>>>(agg, n16);
  k_xl<<<gXl, 256, 0, stream>>>(hB, params + 256, xl, N);
  k_scatter<<<gScat, 256, 0, stream>>>(row, col, deg, xl, agg, E, N);
  k_final<<<gN, 256, 0, stream>>>(agg, gcn2_b, params + 512, (float*)d_out, N);
}